// QuantumEmbeddingLayer_14293651161684
// MI455X (gfx1250) — compile-verified
//
#include <hip/hip_runtime.h>

#define NQ      10
#define SDIM    1024      // 2^NQ
#define CDIM    768
#define NLAYERS 3
#define BATCH   2048

typedef float v2f __attribute__((ext_vector_type(2)));
typedef float v8f __attribute__((ext_vector_type(8)));

__device__ __forceinline__ float2 cmul(float2 a, float2 b) {
    return make_float2(a.x * b.x - a.y * b.y, a.x * b.y + a.y * b.x);
}
__device__ __forceinline__ float2 cadd(float2 a, float2 b) {
    return make_float2(a.x + b.x, a.y + b.y);
}

// ---------------------------------------------------------------------------
// Kernel A: build M = U^T where U is the (batch-independent) unitary of the
// 3 StronglyEntanglingLayers. Row k of M = (U e_k)^T, obtained by simulating
// the layer gates on basis state |k> with the 8KB state held in LDS.
// grid = 1024 (one block per basis state), block = 512 (one thread per pair).
// ---------------------------------------------------------------------------
__global__ void __launch_bounds__(512) build_unitary_kernel(
    const float* __restrict__ weights, float* __restrict__ M_re,
    float* __restrict__ M_im) {
    __shared__ float2 st[SDIM];
    const int tid = threadIdx.x;
    const int k = blockIdx.x;
    st[tid]       = make_float2(tid == k ? 1.f : 0.f, 0.f);
    st[tid + 512] = make_float2((tid + 512) == k ? 1.f : 0.f, 0.f);
    __syncthreads();

    for (int l = 0; l < NLAYERS; ++l) {
        // Rot(phi,theta,omega) = RZ(omega) RY(theta) RZ(phi) on each qubit
        for (int q = 0; q < NQ; ++q) {
            const float* w = weights + (l * NQ + q) * 3;
            const float phi = w[0], theta = w[1], omega = w[2];
            const float ct = cosf(0.5f * theta), sv = sinf(0.5f * theta);
            const float al = 0.5f * (phi + omega), be = 0.5f * (phi - omega);
            const float ca = cosf(al), sa = sinf(al);
            const float cb = cosf(be), sb = sinf(be);
            const float2 g00 = make_float2(ct * ca, -ct * sa);   //  e^{-i a} ct
            const float2 g01 = make_float2(-sv * cb, -sv * sb);  // -e^{+i b} st
            const float2 g10 = make_float2(sv * cb, -sv * sb);   //  e^{-i b} st
            const float2 g11 = make_float2(ct * ca, ct * sa);    //  e^{+i a} ct
            const int mask = 1 << (9 - q);
            const int low = tid & (mask - 1);
            const int high = (tid & ~(mask - 1)) << 1;
            const int i0 = high | low, i1 = i0 | mask;
            const float2 a0 = st[i0], a1 = st[i1];
            const float2 n0 = cadd(cmul(g00, a0), cmul(g01, a1));
            const float2 n1 = cadd(cmul(g10, a0), cmul(g11, a1));
            st[i0] = n0;
            st[i1] = n1;
            __syncthreads();
        }
        // ring of CNOTs, range r = (l % (n-1)) + 1
        const int r = (l % (NQ - 1)) + 1;
        for (int i = 0; i < NQ; ++i) {
            const int c = i, t = (i + r) % NQ;
            const int maskT = 1 << (9 - t), maskC = 1 << (9 - c);
            const int low = tid & (maskT - 1);
            const int high = (tid & ~(maskT - 1)) << 1;
            const int i0 = high | low;
            if (i0 & maskC) {  // control set: swap target-bit pair
                const int i1 = i0 | maskT;
                const float2 a = st[i0], b = st[i1];
                st[i0] = b;
                st[i1] = a;
            }
            __syncthreads();
        }
    }
    M_re[k * SDIM + tid] = st[tid].x;
    M_im[k * SDIM + tid] = st[tid].y;
    M_re[k * SDIM + tid + 512] = st[tid + 512].x;
    M_im[k * SDIM + tid + 512] = st[tid + 512].y;
}

// ---------------------------------------------------------------------------
// Kernel B: angles = z_i @ W_in + b_in, then the embedded product state
// E[b,k] = prod_q (bit ? -i sin(a_q/2) : cos(a_q/2)) = m * (-i)^popcount(k).
// grid = 2048 (one block per sample), block = 256.
// ---------------------------------------------------------------------------
__global__ void __launch_bounds__(256) embed_kernel(
    const float* __restrict__ z, const float* __restrict__ W_in,
    const float* __restrict__ b_in, float* __restrict__ E_re,
    float* __restrict__ E_im) {
    __shared__ float sW[8 * NQ];
    __shared__ float sC[NQ], sS[NQ];
    const int tid = threadIdx.x, b = blockIdx.x;
    const int lane = tid & 31, wv = tid >> 5;

    float acc[NQ];
#pragma unroll
    for (int q = 0; q < NQ; ++q) acc[q] = 0.f;
    for (int d = tid; d < CDIM; d += 256) {
        const float zv = z[b * CDIM + d];
        const float* wr = W_in + d * NQ;
#pragma unroll
        for (int q = 0; q < NQ; ++q) acc[q] += zv * wr[q];
    }
#pragma unroll
    for (int q = 0; q < NQ; ++q) {
        for (int s = 16; s > 0; s >>= 1) acc[q] += __shfl_xor(acc[q], s, 32);
    }
    if (lane == 0) {
#pragma unroll
        for (int q = 0; q < NQ; ++q) sW[wv * NQ + q] = acc[q];
    }
    __syncthreads();
    if (tid < NQ) {
        float a = b_in[tid];
        for (int w = 0; w < 8; ++w) a += sW[w * NQ + tid];
        sC[tid] = cosf(0.5f * a);
        sS[tid] = sinf(0.5f * a);
    }
    __syncthreads();
    for (int i = 0; i < 4; ++i) {
        const int k = tid + i * 256;
        float m = 1.f;
#pragma unroll
        for (int q = 0; q < NQ; ++q) m *= ((k >> (9 - q)) & 1) ? sS[q] : sC[q];
        const int p = __popc(k) & 3;
        E_re[b * SDIM + k] = (p == 0) ? m : (p == 2 ? -m : 0.f);
        E_im[b * SDIM + k] = (p == 1) ? -m : (p == 3 ? m : 0.f);
    }
}

// ---------------------------------------------------------------------------
// Kernel C: complex GEMM F = E @ M via fp32 WMMA (V_WMMA_F32_16X16X4_F32),
// with fused epilogue: p = |F|^2 reduced with +/-1 signs into <Z_w>.
// Each block owns a 16-row batch strip x all 1024 columns, so evs needs no
// global atomics. grid = 128 blocks, block = 256 (8 waves, 8 N-tiles each).
// A layout: lane 0-15 -> M=lane, K={0,1}; lane 16-31 -> K={2,3} (ISA 7.12.2).
// ---------------------------------------------------------------------------
__global__ void __launch_bounds__(256) state_gemm_kernel(
    const float* __restrict__ E_re, const float* __restrict__ E_im,
    const float* __restrict__ M_re, const float* __restrict__ M_im,
    float* __restrict__ evs) {
    __shared__ float sEvs[16 * NQ];
    const int tid = threadIdx.x;
    const int lane = tid & 31, wv = tid >> 5;
    const int m0 = blockIdx.x * 16;
    if (tid < 16 * NQ) sEvs[tid] = 0.f;
    __syncthreads();

    const int half = lane >> 4;  // 0: K pair {0,1}; 1: K pair {2,3}
    const int l15 = lane & 15;
    const float* Ar = E_re + (size_t)(m0 + l15) * SDIM + half * 2;
    const float* Ai = E_im + (size_t)(m0 + l15) * SDIM + half * 2;

    for (int nt = wv; nt < SDIM / 16; nt += 8) {
        const int n0 = nt * 16;
        const float* Br = M_re + (size_t)(half * 2) * SDIM + n0 + l15;
        const float* Bi = M_im + (size_t)(half * 2) * SDIM + n0 + l15;
        v8f rr = {}, ri = {}, ir = {}, ii = {};
#pragma unroll 4
        for (int k0 = 0; k0 < SDIM; k0 += 4) {
            const v2f a_re = *(const v2f*)(Ar + k0);
            const v2f a_im = *(const v2f*)(Ai + k0);
            v2f b_re, b_im;
            b_re.x = Br[(size_t)k0 * SDIM];
            b_re.y = Br[(size_t)(k0 + 1) * SDIM];
            b_im.x = Bi[(size_t)k0 * SDIM];
            b_im.y = Bi[(size_t)(k0 + 1) * SDIM];
            rr = __builtin_amdgcn_wmma_f32_16x16x4_f32(false, a_re, false, b_re,
                                                       (short)0, rr, false, false);
            ii = __builtin_amdgcn_wmma_f32_16x16x4_f32(false, a_im, false, b_im,
                                                       (short)0, ii, false, false);
            ri = __builtin_amdgcn_wmma_f32_16x16x4_f32(false, a_re, false, b_im,
                                                       (short)0, ri, false, false);
            ir = __builtin_amdgcn_wmma_f32_16x16x4_f32(false, a_im, false, b_re,
                                                       (short)0, ir, false, false);
        }
        // C/D layout: VGPR r -> rows {r, r+8}; lanes within half -> N columns
        const int j = n0 + l15;
#pragma unroll
        for (int r = 0; r < 8; ++r) {
            const float fre = rr[r] - ii[r];
            const float fim = ri[r] + ir[r];
            const float p = fre * fre + fim * fim;
            const int rowm = r + half * 8;
#pragma unroll
            for (int w = 0; w < NQ; ++w) {
                float val = ((j >> (9 - w)) & 1) ? -p : p;
                val += __shfl_xor(val, 1, 32);
                val += __shfl_xor(val, 2, 32);
                val += __shfl_xor(val, 4, 32);
                val += __shfl_xor(val, 8, 32);
                if (l15 == 0) atomicAdd(&sEvs[rowm * NQ + w], val);
            }
        }
    }
    __syncthreads();
    if (tid < 16 * NQ) {
        const int row = tid / NQ, w = tid % NQ;
        evs[(size_t)(m0 + row) * NQ + w] = sEvs[tid];
    }
}

// ---------------------------------------------------------------------------
// Kernel D: out = evs @ W_out + b_out (2048x10 @ 10x768). Bandwidth-trivial.
// ---------------------------------------------------------------------------
__global__ void __launch_bounds__(256) out_gemm_kernel(
    const float* __restrict__ evs, const float* __restrict__ W_out,
    const float* __restrict__ b_out, float* __restrict__ out) {
    const int b = blockIdx.x, tid = threadIdx.x;
    float e[NQ];
#pragma unroll
    for (int w = 0; w < NQ; ++w) e[w] = evs[b * NQ + w];
    for (int c = tid; c < CDIM; c += 256) {
        float acc = b_out[c];
#pragma unroll
        for (int w = 0; w < NQ; ++w) acc += e[w] * W_out[w * CDIM + c];
        out[(size_t)b * CDIM + c] = acc;
    }
}

extern "C" void kernel_launch(void* const* d_in, const int* in_sizes, int n_in,
                              void* d_out, int out_size, void* d_ws,
                              size_t ws_size, hipStream_t stream) {
    const float* z_i     = (const float*)d_in[0];
    const float* W_in    = (const float*)d_in[1];
    const float* b_in    = (const float*)d_in[2];
    const float* weights = (const float*)d_in[3];
    const float* W_out   = (const float*)d_in[4];
    const float* b_out   = (const float*)d_in[5];
    float* out = (float*)d_out;

    // workspace: M (2x4MB) + E (2x8MB) + evs (80KB) ~= 25.3 MB
    float* ws   = (float*)d_ws;
    float* M_re = ws;
    float* M_im = M_re + (size_t)SDIM * SDIM;
    float* E_re = M_im + (size_t)SDIM * SDIM;
    float* E_im = E_re + (size_t)BATCH * SDIM;
    float* evs  = E_im + (size_t)BATCH * SDIM;

    build_unitary_kernel<<<SDIM, 512, 0, stream>>>(weights, M_re, M_im);
    embed_kernel<<<BATCH, 256, 0, stream>>>(z_i, W_in, b_in, E_re, E_im);
    state_gemm_kernel<<<BATCH / 16, 256, 0, stream>>>(E_re, E_im, M_re, M_im, evs);
    out_gemm_kernel<<<BATCH, 256, 0, stream>>>(evs, W_out, b_out, out);
}